// RoIAlignAvg_2104533975274
// MI455X (gfx1250) — compile-verified
//
#include <hip/hip_runtime.h>
#include <stdint.h>

// Problem constants (from the reference)
#define C_TOT    256
#define T_SZ     16
#define H_SZ     64
#define W_SZ     64
#define TGRID    5      // TD+1
#define HGRID    8      // AH+1
#define WGRID    8      // AW+1
#define SP_SCALE 0.0625f
#define TM_SCALE 1.0f
#define NTHREADS 128

typedef int v4i __attribute__((ext_vector_type(4)));

// ---------------- CDNA5 async global->LDS (16B / lane) ----------------
#if defined(__HIP_DEVICE_COMPILE__) && defined(__gfx1250__)
__device__ __forceinline__ void async_copy16(void* lds_dst, const void* gsrc) {
#if __has_builtin(__builtin_amdgcn_global_load_async_to_lds_b128)
  __builtin_amdgcn_global_load_async_to_lds_b128(
      (__attribute__((address_space(1))) v4i*)gsrc,
      (__attribute__((address_space(3))) v4i*)lds_dst,
      /*imm offset*/ 0, /*cpol*/ 0);
#else
  uint32_t l = (uint32_t)(uintptr_t)(__attribute__((address_space(3))) void*)lds_dst;
  uint64_t g = (uint64_t)(uintptr_t)gsrc;
  asm volatile("global_load_async_to_lds_b128 %0, %1, off"
               :: "v"(l), "v"(g) : "memory");
#endif
}
__device__ __forceinline__ void async_wait0() {
#if __has_builtin(__builtin_amdgcn_s_wait_asynccnt)
  __builtin_amdgcn_s_wait_asynccnt(0);
#else
  asm volatile("s_wait_asynccnt 0" ::: "memory");
#endif
}
#else
__device__ __forceinline__ void async_copy16(void* lds_dst, const void* gsrc) {
  *(float4*)lds_dst = *(const float4*)gsrc;
}
__device__ __forceinline__ void async_wait0() {}
#endif

// Reference _axis_samples with the per-axis bin size hoisted out (no fp div).
__device__ __forceinline__ void axis_sample(float start, float bin, int i, int size,
                                            int& lo, float& fr, bool& vld) {
  float c   = fmaf(bin, (float)i, start);
  float lof = floorf(c);
  lof = fminf(fmaxf(lof, 0.0f), (float)(size - 2));
  lo  = (int)lof;
  fr  = c - lof;
  vld = (c >= 0.0f) && (c < (float)size);
}

// One block per (roi, channel). Stage feature rows into LDS with coalesced
// 128-bit async copies, trilinear-sample from LDS, fused 2x2x2 avg pool.
__global__ __launch_bounds__(NTHREADS)
void roi_align_avg_kernel(const float* __restrict__ feat,
                          const float* __restrict__ rois,
                          float* __restrict__ out) {
  // rows[tslot=10][hslot=16][w=64] : tslot = 2*tg+dt, hslot = 2*hg+dh
  __shared__ float rows[10 * 16 * 64];                 // 40 KB
  __shared__ float s2[TGRID * HGRID * WGRID];          // sample cube, 1.25 KB

  const int tid = threadIdx.x;
  const int roi = blockIdx.x;
  const int ch  = blockIdx.y;

  const float* rp = rois + (size_t)roi * 7;
  const int   b   = (int)rp[0];
  const float wst = rp[1] * SP_SCALE, wen = rp[4] * SP_SCALE;
  const float hst = rp[2] * SP_SCALE, hen = rp[5] * SP_SCALE;
  const float tst = rp[3] * TM_SCALE, ten = rp[6] * TM_SCALE;

  // Per-axis bin sizes (reciprocal multiply; t-axis /4 is exact)
  const float binW = fmaxf(wen - wst, 0.0f) * (1.0f / (float)(WGRID - 1));
  const float binH = fmaxf(hen - hst, 0.0f) * (1.0f / (float)(HGRID - 1));
  const float binT = fmaxf(ten - tst, 0.0f) * 0.25f;

  // W-span of this ROI (uniform across the block) -> only stage needed float4s
  int w0a; float fdum; bool vdum;
  axis_sample(wst, binW, 0, W_SZ, w0a, fdum, vdum);
  int w0b;
  axis_sample(wst, binW, WGRID - 1, W_SZ, w0b, fdum, vdum);
  const int q0 = w0a >> 2;            // first float4 segment needed
  const int q1 = (w0b + 1) >> 2;      // last  float4 segment needed

  const float* fbase = feat + ((size_t)b * C_TOT + ch) * (size_t)(T_SZ * H_SZ * W_SZ);

  // ---------------- Phase A: stage 160 rows (10 t-slots x 16 h-slots) ------
  // Thread decomposition: lane = tid&15 (float4 within row, loop-invariant),
  // r0 = tid>>4 in 0..7 -> this thread owns h-slots {r0, r0+8} for every t-slot.
  {
    const int lane = tid & 15;
    const int r0   = tid >> 4;
    if (lane >= q0 && lane <= q1) {
      int h0;
      axis_sample(hst, binH, r0 >> 1, H_SZ, h0, fdum, vdum);
      const int hv0 = h0 + (r0 & 1);                 // h value for slot r0
      axis_sample(hst, binH, (r0 + 8) >> 1, H_SZ, h0, fdum, vdum);
      const int hv1 = h0 + (r0 & 1);                 // h value for slot r0+8

      const float* lanebase = fbase + lane * 4;
      float* ldsbase = &rows[r0 * 64 + lane * 4];
      #pragma unroll
      for (int tg = 0; tg < TGRID; ++tg) {
        int t0;
        axis_sample(tst, binT, tg, T_SZ, t0, fdum, vdum);
        #pragma unroll
        for (int dt = 0; dt < 2; ++dt) {
          const int ts = 2 * tg + dt;
          const float* tb = lanebase + (size_t)(t0 + dt) * (H_SZ * W_SZ);
          async_copy16(ldsbase + (ts * 16) * 64,       tb + hv0 * W_SZ);
          async_copy16(ldsbase + (ts * 16 + 8) * 64,   tb + hv1 * W_SZ);
        }
      }
    }
  }
  async_wait0();
  __syncthreads();

  // ---------------- Phase B: trilinear samples from LDS ----------------
  if (tid < HGRID * WGRID) {
    const int hg = tid >> 3;
    const int wg = tid & 7;
    int h0; float fh; bool vh;
    axis_sample(hst, binH, hg, H_SZ, h0, fh, vh);
    int w0; float fw; bool vw;
    axis_sample(wst, binW, wg, W_SZ, w0, fw, vw);
    const float fw0 = 1.0f - fw;

    #pragma unroll
    for (int tg = 0; tg < TGRID; ++tg) {
      int t0; float ft; bool vt;
      axis_sample(tst, binT, tg, T_SZ, t0, ft, vt);
      float acc = 0.0f;
      #pragma unroll
      for (int dt = 0; dt < 2; ++dt) {
        const float wt = dt ? ft : 1.0f - ft;
        #pragma unroll
        for (int dh = 0; dh < 2; ++dh) {
          const float wh  = dh ? fh : 1.0f - fh;
          const float wth = wt * wh;
          const float* rowp = &rows[((2 * tg + dt) * 16 + 2 * hg + dh) * 64 + w0];
          acc += wth * (fw0 * rowp[0] + fw * rowp[1]);
        }
      }
      acc *= (vt && vh && vw) ? 1.0f : 0.0f;   // reference valid mask
      s2[(tg * HGRID + hg) * WGRID + wg] = acc;
    }
  }
  __syncthreads();

  // ---------------- Phase C: fused 2x2x2 avg pool + NT store ----------------
  float* outp = out + ((size_t)roi * C_TOT + ch) * (size_t)(4 * 7 * 7);
  for (int o = tid; o < 4 * 7 * 7; o += NTHREADS) {
    int to  = o / 49;
    int rem = o - to * 49;
    int ho  = rem / 7;
    int wo  = rem - ho * 7;
    float s = 0.0f;
    #pragma unroll
    for (int dt = 0; dt < 2; ++dt)
      #pragma unroll
      for (int dh = 0; dh < 2; ++dh)
        #pragma unroll
        for (int dw = 0; dw < 2; ++dw)
          s += s2[((to + dt) * HGRID + ho + dh) * WGRID + wo + dw];
    // Output has zero reuse: non-temporal store keeps L2 for the feature tensor
    __builtin_nontemporal_store(0.125f * s, &outp[o]);
  }
}

extern "C" void kernel_launch(void* const* d_in, const int* in_sizes, int n_in,
                              void* d_out, int out_size, void* d_ws, size_t ws_size,
                              hipStream_t stream) {
  const float* feat = (const float*)d_in[0];
  const float* rois = (const float*)d_in[1];
  float* out = (float*)d_out;
  const int R = in_sizes[1] / 7;          // 1024
  dim3 grid(R, C_TOT);                    // (roi, channel)
  dim3 block(NTHREADS);
  roi_align_avg_kernel<<<grid, block, 0, stream>>>(feat, rois, out);
}